// ResonantCoupledVortexPDEStep_15994458211424
// MI455X (gfx1250) — compile-verified
//
#include <hip/hip_runtime.h>
#include <math.h>

// Problem constants (match reference)
#define BB 8
#define CC 32
#define HH 512
#define WW 512
#define PLANES (BB * CC)          // 256
#define PLANE_ELEMS (HH * WW)     // 262144
#define DT 0.1f

#define TILE_H 16
#define NCHUNK 32                 // reduction chunks per plane

typedef int   v4i __attribute__((ext_vector_type(4)));
typedef float v4f __attribute__((ext_vector_type(4)));
typedef __attribute__((address_space(1))) v4i* gv4i_ptr;
typedef __attribute__((address_space(3))) v4i* lv4i_ptr;

// ---------------- async global->LDS helpers (CDNA5 path) ----------------

__device__ __forceinline__ void async_copy_b128(const float* gsrc, float* ldst) {
#if __has_builtin(__builtin_amdgcn_global_load_async_to_lds_b128)
    __builtin_amdgcn_global_load_async_to_lds_b128(
        (gv4i_ptr)(gsrc),
        (lv4i_ptr)(ldst),
        /*imm offset*/0, /*cpol*/0);
#else
    *(v4f*)ldst = *(const v4f*)gsrc;         // compile-safe fallback
#endif
}

__device__ __forceinline__ void async_wait_zero() {
#if __has_builtin(__builtin_amdgcn_s_wait_asynccnt)
    __builtin_amdgcn_s_wait_asynccnt(0);
#elif __has_builtin(__builtin_amdgcn_global_load_async_to_lds_b128)
    asm volatile("s_wait_asynccnt 0" ::: "memory");
#endif
}

// ---------------- pass 3 FIRST in TU (so disasm snippet shows it) --------
// Stencil update with async LDS row-tile staging.

__device__ __forceinline__ float pde_point(float e, float l, float r, float u, float d,
                                           float f, float nuc, float phi, float famp) {
    const float lap = u + d + l + r - 4.0f * e;
    const float gx  = 0.5f * (r - l);
    const float gy  = 0.5f * (d - u);
    const float dS  = fmaf(nuc, lap, fmaf(-e, gx + gy, fmaf(phi, e, famp * f)));
    return fmaf(DT, dS, e);
}

__global__ __launch_bounds__(256)
void k_step(const float* __restrict__ S, const float* __restrict__ F,
            const float* __restrict__ nu, const float* __restrict__ phi_arr,
            const float* __restrict__ famp_p, float* __restrict__ out) {
    const int plane = blockIdx.y;            // 0..255
    const int ty    = blockIdx.x;            // 0..31 (row tiles)
    const int t     = threadIdx.x;           // 0..255
    const int c     = plane & (CC - 1);
    const size_t pbase = (size_t)plane * PLANE_ELEMS;
    const int h0 = ty * TILE_H;

    // full-width rows: periodic column wrap is just an LDS index mask
    __shared__ float tile[TILE_H + 2][WW];   // 18 * 512 * 4 = 36 KB

    // Stage 18 rows via async b128 copies: 2304 chunks / 256 threads = 9 each.
#pragma unroll
    for (int k = 0; k < 9; ++k) {
        const int i  = t + 256 * k;          // chunk id 0..2303
        const int r  = i >> 7;               // row in tile (128 chunks/row)
        const int c4 = (i & 127) << 2;       // starting column
        const int g  = (h0 - 1 + r) & (HH - 1);
        async_copy_b128(S + pbase + (size_t)g * WW + c4, &tile[r][c4]);
    }
    async_wait_zero();
    __syncthreads();

    const float nuc  = nu[c];
    const float phi  = phi_arr[plane];
    const float famp = famp_p[0];

    // 16 rows * 128 float4-chunks = 2048 chunks / 256 threads = 8 each.
#pragma unroll
    for (int k = 0; k < 8; ++k) {
        const int i   = t + 256 * k;
        const int row = i >> 7;              // 0..15
        const int c4  = (i & 127) << 2;

        const v4f ctr = *(const v4f*)&tile[row + 1][c4];
        const v4f up  = *(const v4f*)&tile[row + 0][c4];
        const v4f dn  = *(const v4f*)&tile[row + 2][c4];
        const float lft = tile[row + 1][(c4 - 1) & (WW - 1)];
        const float rgt = tile[row + 1][(c4 + 4) & (WW - 1)];

        const size_t gidx = pbase + (size_t)(h0 + row) * WW + c4;
        // forcing is a pure stream: non-temporal load keeps L2 free for S
        const v4f f = __builtin_nontemporal_load((const v4f*)(F + gidx));

        v4f o;
        o.x = pde_point(ctr.x, lft,   ctr.y, up.x, dn.x, f.x, nuc, phi, famp);
        o.y = pde_point(ctr.y, ctr.x, ctr.z, up.y, dn.y, f.y, nuc, phi, famp);
        o.z = pde_point(ctr.z, ctr.y, ctr.w, up.z, dn.z, f.z, nuc, phi, famp);
        o.w = pde_point(ctr.w, ctr.z, rgt,   up.w, dn.w, f.w, nuc, phi, famp);

        // output is write-once: non-temporal store, don't pollute L2
        __builtin_nontemporal_store(o, (v4f*)(out + gidx));
    }
}

// ---------------- pass 1: per-plane sum of squares (deterministic) -------

__global__ __launch_bounds__(256)
void k_partial_sumsq(const float* __restrict__ S, float* __restrict__ partial) {
    const int plane = blockIdx.y;        // 0..255
    const int chunk = blockIdx.x;        // 0..31
    const int t     = threadIdx.x;       // 0..255

    const size_t base = (size_t)plane * PLANE_ELEMS + (size_t)chunk * (PLANE_ELEMS / NCHUNK);
    const v4f* S4 = (const v4f*)(S + base);

    float acc = 0.0f;
#pragma unroll
    for (int k = 0; k < 8; ++k) {        // 8 * 256 float4 = 8192 elems
        v4f v = S4[t + 256 * k];
        acc = fmaf(v.x, v.x, acc);
        acc = fmaf(v.y, v.y, acc);
        acc = fmaf(v.z, v.z, acc);
        acc = fmaf(v.w, v.w, acc);
    }

    __shared__ float red[256];
    red[t] = acc;
    __syncthreads();
#pragma unroll
    for (int s = 128; s > 0; s >>= 1) {
        if (t < s) red[t] += red[t + s];
        __syncthreads();
    }
    if (t == 0) partial[plane * NCHUNK + chunk] = red[0];
}

// ---------------- pass 2: fold partials -> phi[plane] --------------------

__global__ __launch_bounds__(256)
void k_phi(const float* __restrict__ partial, const float* __restrict__ cs,
           float* __restrict__ phi) {
    const int p = threadIdx.x;           // plane id, one block of 256
    float s = 0.0f;
#pragma unroll
    for (int j = 0; j < NCHUNK; ++j) s += partial[p * NCHUNK + j];
    const float A  = sqrtf(s);
    const float Sn = cs[p & (CC - 1)] / (A + 1e-8f);
    phi[p] = 0.05f * tanhf(0.05f * Sn);
}

// ---------------- launch ----------------

extern "C" void kernel_launch(void* const* d_in, const int* in_sizes, int n_in,
                              void* d_out, int out_size, void* d_ws, size_t ws_size,
                              hipStream_t stream) {
    const float* S    = (const float*)d_in[0];
    const float* F    = (const float*)d_in[1];
    const float* nu   = (const float*)d_in[2];
    const float* cs   = (const float*)d_in[3];
    const float* famp = (const float*)d_in[4];
    float* out = (float*)d_out;

    float* partial = (float*)d_ws;                    // 256*32 floats
    float* phi     = partial + PLANES * NCHUNK;       // 256 floats

    dim3 blk(256);
    dim3 grid_red(NCHUNK, PLANES);                    // 32 x 256
    dim3 grid_step(HH / TILE_H, PLANES);              // 32 x 256

    k_partial_sumsq<<<grid_red, blk, 0, stream>>>(S, partial);
    k_phi<<<dim3(1), blk, 0, stream>>>(partial, cs, phi);
    k_step<<<grid_step, blk, 0, stream>>>(S, F, nu, phi, famp, out);
}